// localAE_69320772157915
// MI455X (gfx1250) — compile-verified
//
#include <hip/hip_runtime.h>
#include <stddef.h>

#define N_NODES 100000

typedef float v2f __attribute__((ext_vector_type(2)));
typedef float v8f __attribute__((ext_vector_type(8)));

// ---------------------------------------------------------------------------
// Degree computation: deg[i] = 1 (self loop) + #incoming edges, then rsqrt.
// ---------------------------------------------------------------------------
__global__ void deg_init_kernel(float* __restrict__ deg, int n) {
  int i = blockIdx.x * blockDim.x + threadIdx.x;
  if (i < n) deg[i] = 1.0f;  // self loop contributes 1
}

__global__ void deg_accum_kernel(const long long* __restrict__ dst,
                                 float* __restrict__ deg, int nE) {
  int e = blockIdx.x * blockDim.x + threadIdx.x;
  if (e < nE) {
    int d = (int)dst[e];
    unsafeAtomicAdd(&deg[d], 1.0f);  // global_atomic_add_f32
  }
}

__global__ void dinv_kernel(float* __restrict__ deg, int n) {
  int i = blockIdx.x * blockDim.x + threadIdx.x;
  if (i < n) deg[i] = rsqrtf(deg[i]);  // deg >= 1 always (self loop)
}

// ---------------------------------------------------------------------------
// f32 WMMA GEMM: C[M, NT*16] = A[M,K] * B[K, NT*16], row-major, M % 16 == 0.
// One wave per 16-row block; the A fragment (2 VGPRs) is loaded ONCE and
// reused across all NT column tiles (NT v8f accumulators), eliminating the
// 4x/8x redundant A traffic. V_WMMA_F32_16X16X4_F32 over K in steps of 4.
//   A 16x4: lane = m + 16*kh holds A[m][k0+2*kh + {0,1}] in {v0,v1}.
//   B 4x16: lane = n + 16*kh holds B[k0+2*kh + {0,1}][n] in {v0,v1}.
//   C 16x16: lane = n + 16*g, VGPR i holds C[8*g + i][n].
// ---------------------------------------------------------------------------
template <int K, int NT>
__global__ __launch_bounds__(256) void wmma_gemm_f32(
    const float* __restrict__ A, const float* __restrict__ B,
    float* __restrict__ C, int M) {
  constexpr int Nout = NT * 16;
  const int lane = threadIdx.x;                       // 0..31 (wave32)
  const int tile = blockIdx.x * blockDim.y + threadIdx.y;  // row-tile (uniform)
  if (tile >= (M >> 4)) return;                       // whole-wave exit only

  const int mn = lane & 15;   // M for A-loads, N for B-loads / C-stores
  const int kh = lane >> 4;   // K-half selector (0 or 1)

  const float* aptr  = A + (size_t)(tile * 16 + mn) * K + 2 * kh;  // 8B aligned
  const float* bbase = B + (size_t)(2 * kh) * Nout + mn;

  v8f acc[NT];
#pragma unroll
  for (int j = 0; j < NT; ++j) acc[j] = v8f{};

#pragma unroll 4
  for (int k0 = 0; k0 < K; k0 += 4) {
    v2f a = *(const v2f*)(aptr + k0);
#pragma unroll
    for (int j = 0; j < NT; ++j) {
      v2f b;
      b.x = bbase[(size_t)k0 * Nout + j * 16];
      b.y = bbase[(size_t)(k0 + 1) * Nout + j * 16];
      acc[j] = __builtin_amdgcn_wmma_f32_16x16x4_f32(
          /*neg_a=*/false, a, /*neg_b=*/false, b,
          /*c_mod=*/(short)0, acc[j], /*reuse_a=*/false, /*reuse_b=*/false);
    }
  }

  float* cp = C + (size_t)(tile * 16 + kh * 8) * Nout + mn;
#pragma unroll
  for (int j = 0; j < NT; ++j)
#pragma unroll
    for (int i = 0; i < 8; ++i) cp[(size_t)i * Nout + j * 16] = acc[j][i];
}

// ---------------------------------------------------------------------------
// out[i][f] = bias[f] + h[i][f] * dinv[i]^2   (bias + self-loop term),
// 4 channels per thread (b128 load/store).
// ---------------------------------------------------------------------------
template <int F>
__global__ void out_init_kernel(const float* __restrict__ h,
                                const float* __restrict__ dinv,
                                const float* __restrict__ bias,
                                float* __restrict__ out, int n) {
  constexpr int Q = F / 4;  // float4 groups per row
  int idx = blockIdx.x * blockDim.x + threadIdx.x;
  if (idx >= n * Q) return;
  int i = idx / Q;
  int q = idx - i * Q;
  float di = dinv[i];
  float s = di * di;
  const float4 hv = *(const float4*)&h[(size_t)i * F + q * 4];
  const float4 bv = *(const float4*)&bias[q * 4];
  float4 o;
  o.x = bv.x + hv.x * s;
  o.y = bv.y + hv.y * s;
  o.z = bv.z + hv.z * s;
  o.w = bv.w + hv.w * s;
  *(float4*)&out[(size_t)i * F + q * 4] = o;
}

// ---------------------------------------------------------------------------
// Edge scatter: out[dst] += h[src] * dinv[src]*dinv[dst].
// One thread per (edge, 4 channels): one b128 gather + 4 f32 atomics; the
// whole feature matrix (25.6/51.2 MB) lives in the 192 MB L2, so gathers and
// atomics stay on-die.
// ---------------------------------------------------------------------------
template <int F>
__global__ void scatter_kernel(const long long* __restrict__ src,
                               const long long* __restrict__ dst,
                               const float* __restrict__ h,
                               const float* __restrict__ dinv,
                               float* __restrict__ out, int nE) {
  constexpr int Q = F / 4;  // threads per edge
  int idx = blockIdx.x * blockDim.x + threadIdx.x;
  if (idx >= nE * Q) return;
  int e = idx / Q;
  int q = idx - e * Q;
  int s = (int)src[e];
  int d = (int)dst[e];
  float norm = dinv[s] * dinv[d];
  const float4 hv = *(const float4*)&h[(size_t)s * F + q * 4];
  float* op = &out[(size_t)d * F + q * 4];
  unsafeAtomicAdd(op + 0, hv.x * norm);
  unsafeAtomicAdd(op + 1, hv.y * norm);
  unsafeAtomicAdd(op + 2, hv.z * norm);
  unsafeAtomicAdd(op + 3, hv.w * norm);
}

// ---------------------------------------------------------------------------
// Launch
// ---------------------------------------------------------------------------
extern "C" void kernel_launch(void* const* d_in, const int* in_sizes, int n_in,
                              void* d_out, int out_size, void* d_ws,
                              size_t ws_size, hipStream_t stream) {
  const float* x        = (const float*)d_in[0];       // [N,128]
  const long long* edge = (const long long*)d_in[1];   // [2,E] int64
  const float* W_enc    = (const float*)d_in[2];       // [128,64]
  const float* b_enc    = (const float*)d_in[3];       // [64]
  const float* W_dec    = (const float*)d_in[4];       // [64,128]
  const float* b_dec    = (const float*)d_in[5];       // [128]

  const int N  = N_NODES;
  const int nE = in_sizes[1] / 2;
  const long long* srcA = edge;
  const long long* dstA = edge + nE;

  float* out1 = (float*)d_out;           // loc_emb         [N,64]
  float* out2 = out1 + (size_t)N * 64;   // loc_reconstruct [N,128]

  // Workspace: dinv[N] | h1[N,64] | h2[N,128]  (~77.2 MB)
  char* ws = (char*)d_ws;
  const size_t o1 = ((size_t)N * sizeof(float) + 255) & ~(size_t)255;
  const size_t o2 = o1 + (size_t)N * 64 * sizeof(float);
  float* dinv = (float*)ws;
  float* h1   = (float*)(ws + o1);
  float* h2   = (float*)(ws + o2);

  const int T = 256;
  const int rowTiles = N / 16;  // 6250, N % 16 == 0

  // 1) degrees -> dinv
  deg_init_kernel<<<(N + T - 1) / T, T, 0, stream>>>(dinv, N);
  deg_accum_kernel<<<(nE + T - 1) / T, T, 0, stream>>>(dstA, dinv, nE);
  dinv_kernel<<<(N + T - 1) / T, T, 0, stream>>>(dinv, N);

  // 2) layer 1: h1 = x @ W_enc  (K=128, Nout=64) -> init + scatter into out1
  {
    wmma_gemm_f32<128, 4><<<(rowTiles + 7) / 8, dim3(32, 8), 0, stream>>>(
        x, W_enc, h1, N);
    int itotal = N * (64 / 4);
    out_init_kernel<64><<<(itotal + T - 1) / T, T, 0, stream>>>(h1, dinv,
                                                                b_enc, out1, N);
    int etotal = nE * (64 / 4);
    scatter_kernel<64><<<(etotal + T - 1) / T, T, 0, stream>>>(
        srcA, dstA, h1, dinv, out1, nE);
  }

  // 3) layer 2: h2 = out1 @ W_dec  (K=64, Nout=128) -> init + scatter into out2
  {
    wmma_gemm_f32<64, 8><<<(rowTiles + 7) / 8, dim3(32, 8), 0, stream>>>(
        out1, W_dec, h2, N);
    int itotal = N * (128 / 4);
    out_init_kernel<128><<<(itotal + T - 1) / T, T, 0, stream>>>(
        h2, dinv, b_dec, out2, N);
    int etotal = nE * (128 / 4);
    scatter_kernel<128><<<(etotal + T - 1) / T, T, 0, stream>>>(
        srcA, dstA, h2, dinv, out2, nE);
  }
}